// Model14_9620726743227
// MI455X (gfx1250) — compile-verified
//
#include <hip/hip_runtime.h>
#include <hip/hip_bf16.h>
#include <math.h>

// ---------------------------------------------------------------------------
// MI455X (gfx1250) implementation of the GNN reference.
//
// Roofline: ~10 GFLOP dense GEMM vs ~0.6 GB of gather traffic -> memory
// bound at 23.3 TB/s (~25us). All dense matmuls use the exact-f32 matrix op
// V_WMMA_F32_16X16X4_F32 (wave32, 16x16 tile, K=4): full reference precision
// on the WMMA pipe. Weights live in a zero-padded, TRANSPOSED LDS tile
// [Cpad][Kpad] (Cpad a template constant) so each B fragment is one aligned
// ds_load_b64 and the hot loop has no multiplies in its address math.
//
// Param flattening assumption (jax.tree_util.tree_flatten: sorted dict keys,
// tuples in order), d_in[16..79]:
//   16 at.W 17 at.b 18 at2.W 19 at2.b
//   20 att.Wg1 21 att.Wg2 22 att.Wn1 23 att.Wn2 24 bg1 25 bg2 26 bn1 27 bn2
//   28 dt.W 29 dt.b 30 dt2.W 31 dt2.b 32 fol.W 33 fol.b
//   34+9l: g.Wbeta, g.Wk, g.Wq, g.Ws, g.Wv, g.bk, g.bq, g.bs, g.bv
//   61 lin1.W 62 lin1.b 63 lin2.W 64 lin2.b
//   65+3l: n.bias, n.mean_scale, n.weight
//   74 oa.W 75 oa.b 76 tt.W 77 tt.b 78 tt2.W 79 tt2.b
// ---------------------------------------------------------------------------

#define GDIM 50
#define NPG  64
#define NM   32
#define EPSN 1e-5f
#define INV_SQRT_D 0.14142135623730951f  // 1/sqrt(50)

typedef float v2f __attribute__((ext_vector_type(2)));
typedef float v8f __attribute__((ext_vector_type(8)));

__device__ __forceinline__ v8f wmma4(v2f a, v2f b, v8f c) {
#if __has_builtin(__builtin_amdgcn_wmma_f32_16x16x4_f32)
  return __builtin_amdgcn_wmma_f32_16x16x4_f32(false, a, false, b, (short)0, c,
                                               false, false);
#else
  asm volatile("v_wmma_f32_16x16x4_f32 %0, %1, %2, %0"
               : "+v"(c) : "v"(a), "v"(b));
  return c;
#endif
}

__device__ __forceinline__ float apply_act(float y, int act) {
  if (act == 1) return fmaxf(y, 0.f);
  if (act == 2) return (y > 0.f) ? y : 0.01f * y;
  return y;
}

// Stage W[K,C] into LDS transposed + zero-padded: sW[c*Kpad + k].
// Kpad multiple of 4 -> each column 16B aligned; k0+2*hi even -> b64 loads OK.
template <int CPAD>
__device__ __forceinline__ void stage_weights_t(float* sW, const float* W,
                                                int K, int C, int Kpad) {
  for (int i = threadIdx.x; i < CPAD * Kpad; i += blockDim.x) {
    const int c = i / Kpad, k = i - c * Kpad;
    sW[i] = (k < K && c < C) ? W[k * C + c] : 0.f;
  }
  __syncthreads();
}

// ---------------------------------------------------------------------------
// Generic GEMM: Y[row, col] = act( X[row, 0:K] @ W[K,C] + bias[col] )
// grid.x = rows/16 (rows multiple of 16), blockDim = 32 * (CPAD/16).
// Each wave owns one 16-col tile.
// ---------------------------------------------------------------------------
template <int CPAD>
__global__ void k_gemm(const float* __restrict__ X, int ldx, int K,
                       const float* __restrict__ W, int C,
                       const float* __restrict__ bias,
                       float* __restrict__ Y, int ldy, int act) {
  extern __shared__ float sW[];
  const int Kpad = (K + 3) & ~3;
  stage_weights_t<CPAD>(sW, W, K, C, Kpad);

  const int tid = threadIdx.x;
  const int lane = tid & 31;
  const int col0 = (tid >> 5) << 4;
  const int m = lane & 15;
  const int hi = lane >> 4;               // 0: K+{0,1}, 1: K+{2,3}
  const int col = col0 + m;
  const long row0 = (long)blockIdx.x << 4;
  const float* Xrow = X + (row0 + m) * (long)ldx;
  const float* sWcol = sW + col * Kpad;   // transposed: contiguous in k
  __builtin_prefetch(Xrow, 0, 0);         // global_prefetch_b8

  v8f acc;
  const float bv = (col < C && bias) ? bias[col] : 0.f;
#pragma unroll
  for (int r = 0; r < 8; ++r) acc[r] = bv;

  const int Kmain = K & ~3;
  int k0 = 0;
  for (; k0 < Kmain; k0 += 4) {           // branch-free hot loop
    const int ka = k0 + (hi << 1);        // even -> 8B-aligned LDS b64
    v2f a, b;
    a[0] = Xrow[ka];
    a[1] = Xrow[ka + 1];
    b = *(const v2f*)(sWcol + ka);        // one ds_load_b64
    acc = wmma4(a, b, acc);
  }
  if (k0 < K) {                           // guarded tail (K % 4 != 0)
    const int ka = k0 + (hi << 1);
    v2f a, b;
    a[0] = (ka     < K) ? Xrow[ka]     : 0.f;
    a[1] = (ka + 1 < K) ? Xrow[ka + 1] : 0.f;
    b = *(const v2f*)(sWcol + ka);        // zero-padded beyond K
    acc = wmma4(a, b, acc);
  }

  if (col < C) {
#pragma unroll
    for (int r = 0; r < 8; ++r) {
      const long row = row0 + r + (hi << 3);
      Y[row * (long)ldy + col] = apply_act(acc[r], act);
    }
  }
}

// GEMM whose A rows are assembled on the fly:
//   [ D0[row](w0) | XF[idxA[row]](165) | XF[idxB[row]](165) ]
// Segment selection is wave-uniform per 4-wide K block (scalar branches);
// only blocks straddling a segment edge take the per-element path.
template <int CPAD>
__global__ void k_gemm_gather(const float* __restrict__ D0, int w0,
                              const int* __restrict__ idxA,
                              const int* __restrict__ idxB,
                              const float* __restrict__ XF, int K,
                              const float* __restrict__ W, int C,
                              const float* __restrict__ bias,
                              float* __restrict__ Y, int ldy, int act) {
  extern __shared__ float sW[];
  const int Kpad = (K + 3) & ~3;
  stage_weights_t<CPAD>(sW, W, K, C, Kpad);

  const int tid = threadIdx.x;
  const int lane = tid & 31;
  const int col0 = (tid >> 5) << 4;
  const int m = lane & 15;
  const int hi = lane >> 4;
  const int col = col0 + m;
  const long row0 = (long)blockIdx.x << 4;
  const long row = row0 + m;
  const float* pD = D0 + row * (long)w0;
  const float* pA = XF + (long)idxA[row] * 165;
  const float* pB = idxB ? (XF + (long)idxB[row] * 165) : pA;
  const int w1 = w0 + 165;
  const float* sWcol = sW + col * Kpad;
  __builtin_prefetch(pA, 0, 0);

  auto fetchf = [&](int k) -> float {
    if (k >= K) return 0.f;
    if (k < w0) return pD[k];
    if (k < w1) return pA[k - w0];
    return pB[k - w1];
  };

  v8f acc;
  const float bv = (col < C && bias) ? bias[col] : 0.f;
#pragma unroll
  for (int r = 0; r < 8; ++r) acc[r] = bv;

  const int Kmain = K & ~3;
  int k0 = 0;
  for (; k0 < Kmain; k0 += 4) {
    const int ka = k0 + (hi << 1);
    v2f a, b;
    if (k0 + 3 < w0) {                       // uniform: whole block in D0
      a[0] = pD[ka];
      a[1] = pD[ka + 1];
    } else if (k0 >= w0 && k0 + 3 < w1) {    // uniform: whole block in XF[idxA]
      a[0] = pA[ka - w0];
      a[1] = pA[ka - w0 + 1];
    } else if (k0 >= w1) {                   // uniform: whole block in XF[idxB]
      a[0] = pB[ka - w1];
      a[1] = pB[ka - w1 + 1];
    } else {                                 // boundary block (<=3 of these)
      a[0] = fetchf(ka);
      a[1] = fetchf(ka + 1);
    }
    b = *(const v2f*)(sWcol + ka);
    acc = wmma4(a, b, acc);
  }
  if (k0 < K) {
    const int ka = k0 + (hi << 1);
    v2f a, b;
    a[0] = fetchf(ka);
    a[1] = fetchf(ka + 1);
    b = *(const v2f*)(sWcol + ka);
    acc = wmma4(a, b, acc);
  }

  if (col < C) {
#pragma unroll
    for (int r = 0; r < 8; ++r) {
      const long row2 = row0 + r + (hi << 3);
      Y[row2 * (long)ldy + col] = apply_act(acc[r], act);
    }
  }
}

// ------------------------------ helpers ------------------------------------
__global__ void k_fill(float* p, float v, long n) {
  long i = (long)blockIdx.x * blockDim.x + threadIdx.x;
  if (i < n) p[i] = v;
}

__global__ void k_copy_x1(const float* __restrict__ x1, float* __restrict__ xf,
                          long n15) {
  long i = (long)blockIdx.x * blockDim.x + threadIdx.x;
  if (i < n15) xf[(i / 15) * 165 + (i % 15)] = x1[i];
}

// monotone uint key for float atomic max
__device__ __forceinline__ unsigned fkey(float f) {
  unsigned b = __float_as_uint(f);
  return (b & 0x80000000u) ? ~b : (b | 0x80000000u);
}
__device__ __forceinline__ float funkey(unsigned k) {
  return __uint_as_float((k & 0x80000000u) ? (k & 0x7FFFFFFFu) : ~k);
}

__global__ void k_edge_logits(const int* __restrict__ src,
                              const int* __restrict__ dst,
                              const float* __restrict__ q,
                              const float* __restrict__ kk,
                              float* __restrict__ logits,
                              unsigned* __restrict__ nmax, long E) {
  long e = (long)blockIdx.x * blockDim.x + threadIdx.x;
  if (e >= E) return;
  const long s = src[e], d = dst[e];
  float acc = 0.f;
  for (int c = 0; c < GDIM; ++c) acc += q[d * GDIM + c] * kk[s * GDIM + c];
  const float l = acc * INV_SQRT_D;
  logits[e] = l;
  atomicMax(&nmax[d], fkey(l));
}

__global__ void k_edge_exp(const int* __restrict__ dst,
                           const float* __restrict__ logits,
                           const unsigned* __restrict__ nmax,
                           float* __restrict__ expv,
                           float* __restrict__ nsum, long E) {
  long e = (long)blockIdx.x * blockDim.x + threadIdx.x;
  if (e >= E) return;
  const long d = dst[e];
  const float ex = expf(logits[e] - funkey(nmax[d]));
  expv[e] = ex;
  atomicAdd(&nsum[d], ex);
}

__global__ void k_edge_agg(const int* __restrict__ src,
                           const int* __restrict__ dst,
                           const float* __restrict__ expv,
                           const float* __restrict__ nsum,
                           const float* __restrict__ v,
                           float* __restrict__ agg, long E) {
  long e = (long)blockIdx.x * blockDim.x + threadIdx.x;
  if (e >= E) return;
  const long s = src[e], d = dst[e];
  const float w = expv[e] / nsum[d];
  for (int c = 0; c < GDIM; ++c)
    atomicAdd(&agg[d * GDIM + c], w * v[s * GDIM + c]);
}

// beta-gated skip combine + relu
__global__ void k_beta(const float* __restrict__ skip,
                       const float* __restrict__ agg,
                       const float* __restrict__ Wb,  // [150,1]
                       float* __restrict__ out, long N) {
  long i = (long)blockIdx.x * blockDim.x + threadIdx.x;
  if (i >= N) return;
  float t = 0.f;
  for (int c = 0; c < GDIM; ++c) {
    const float sk = skip[i * GDIM + c], ag = agg[i * GDIM + c];
    t += Wb[c] * sk + Wb[GDIM + c] * ag + Wb[2 * GDIM + c] * (sk - ag);
  }
  const float beta = 1.f / (1.f + expf(-t));
  for (int c = 0; c < GDIM; ++c) {
    const float sk = skip[i * GDIM + c], ag = agg[i * GDIM + c];
    out[i * GDIM + c] = fmaxf(beta * sk + (1.f - beta) * ag, 0.f);
  }
}

// GraphNorm over 64 contiguous nodes/graph; writes into xfull column slice.
__global__ void k_gnorm(const float* __restrict__ xin,
                        const float* __restrict__ wgt,
                        const float* __restrict__ bias,
                        const float* __restrict__ ms,
                        float* __restrict__ xf, int colOff) {
  __shared__ float tile[NPG * GDIM];
  __shared__ float mm[GDIM];
  __shared__ float rs[GDIM];
  const int g = blockIdx.x, tid = threadIdx.x;
  const long base = (long)g * NPG * GDIM;
  for (int i = tid; i < NPG * GDIM; i += blockDim.x) tile[i] = xin[base + i];
  __syncthreads();
  if (tid < GDIM) {
    float s = 0.f;
    for (int n = 0; n < NPG; ++n) s += tile[n * GDIM + tid];
    mm[tid] = ms[tid] * (s * (1.f / NPG));
  }
  __syncthreads();
  if (tid < GDIM) {
    float v = 0.f;
    const float c0 = mm[tid];
    for (int n = 0; n < NPG; ++n) {
      const float d = tile[n * GDIM + tid] - c0;
      v += d * d;
    }
    rs[tid] = wgt[tid] * rsqrtf(v * (1.f / NPG) + EPSN);
  }
  __syncthreads();
  for (int i = tid; i < NPG * GDIM; i += blockDim.x) {
    const int n = i / GDIM, c = i % GDIM;
    xf[((long)g * NPG + n) * 165 + colOff + c] =
        (tile[i] - mm[c]) * rs[c] + bias[c];
  }
}

__global__ void k_gate(const float* __restrict__ hn1,
                       const float* __restrict__ Wg2,
                       const float* __restrict__ bg2,
                       float* __restrict__ gl, long N) {
  long i = (long)blockIdx.x * blockDim.x + threadIdx.x;
  if (i >= N) return;
  float acc = bg2[0];
  for (int c = 0; c < GDIM; ++c) acc += hn1[i * GDIM + c] * Wg2[c];
  gl[i] = acc;
}

__global__ void k_gate_pool(const float* __restrict__ gl,
                            const float* __restrict__ hn,
                            float* __restrict__ pooledRelu) {
  __shared__ float red[NPG];
  __shared__ float sg[NPG];
  const int g = blockIdx.x, tid = threadIdx.x;
  const float x = gl[(long)g * NPG + tid];
  red[tid] = x; __syncthreads();
  for (int s = 32; s > 0; s >>= 1) {
    if (tid < s) red[tid] = fmaxf(red[tid], red[tid + s]);
    __syncthreads();
  }
  const float mx = red[0]; __syncthreads();
  const float e = expf(x - mx);
  red[tid] = e; __syncthreads();
  for (int s = 32; s > 0; s >>= 1) {
    if (tid < s) red[tid] += red[tid + s];
    __syncthreads();
  }
  sg[tid] = e / red[0]; __syncthreads();
  if (tid < GDIM) {
    float a = 0.f;
    for (int n = 0; n < NPG; ++n)
      a += sg[n] * hn[((long)g * NPG + n) * GDIM + tid];
    pooledRelu[g * GDIM + tid] = fmaxf(a, 0.f);
  }
}

__global__ void k_value(const float* __restrict__ pooled,
                        const float* __restrict__ x2,
                        const float* __restrict__ W1,
                        const float* __restrict__ b1,
                        const float* __restrict__ W2,
                        const float* __restrict__ b2,
                        float* __restrict__ out) {
  __shared__ float hid[GDIM];
  const int g = blockIdx.x, j = threadIdx.x;
  if (j < GDIM) {
    float acc = b1[j];
    for (int kk = 0; kk < GDIM; ++kk)
      acc += pooled[g * GDIM + kk] * W1[kk * GDIM + j];
    for (int kk = 0; kk < 4; ++kk)
      acc += x2[g * 4 + kk] * W1[(GDIM + kk) * GDIM + j];
    hid[j] = fmaxf(acc, 0.f);
  }
  __syncthreads();
  if (j == 0) {
    float v = b2[0];
    for (int c = 0; c < GDIM; ++c) v += hid[c] * W2[c];
    out[g] = tanhf(v);
  }
}

__global__ void k_scatter(const float* __restrict__ ordersO,
                          const int* __restrict__ ab,
                          const int* __restrict__ tb,
                          const int* __restrict__ db, long NORD,
                          float* __restrict__ tmp) {
  long i = (long)blockIdx.x * blockDim.x + threadIdx.x;
  if (i >= 3 * NORD) return;
  const long r = i / NORD, loc = i - r * NORD;
  const int b = (r == 0) ? ab[loc] : (r == 1) ? tb[loc] : db[loc];
  for (int c = 0; c < 20; ++c)
    atomicAdd(&tmp[(long)b * 20 + c], ordersO[i * 20 + c]);
}

__global__ void k_fol_lsm(const float* __restrict__ tmp,
                          const float* __restrict__ Wf,
                          const float* __restrict__ bf,
                          float* __restrict__ out) {
  __shared__ float red[NM];
  const int g = blockIdx.x, m = threadIdx.x;
  const long row = (long)g * NM + m;
  float l = bf[0];
  for (int c = 0; c < 20; ++c) l += fmaxf(tmp[row * 20 + c], 0.f) * Wf[c];
  red[m] = l; __syncthreads();
  for (int s = 16; s > 0; s >>= 1) {
    if (m < s) red[m] = fmaxf(red[m], red[m + s]);
    __syncthreads();
  }
  const float mx = red[0]; __syncthreads();
  red[m] = expf(l - mx); __syncthreads();
  for (int s = 16; s > 0; s >>= 1) {
    if (m < s) red[m] += red[m + s];
    __syncthreads();
  }
  out[row] = l - mx - logf(red[0]);
}

// ---------------------------------------------------------------------------
static inline size_t shmem_bytes(int K, int C) {
  const int Cpad = (C + 15) & ~15;
  const int Kpad = (K + 3) & ~3;
  return (size_t)Kpad * Cpad * sizeof(float);
}

static inline void gemm(hipStream_t s, const float* X, int ldx, int K,
                        long rows, const float* W, int C, const float* b,
                        float* Y, int ldy, int act) {
  const int Cpad = (C + 15) & ~15;
  const dim3 grid((unsigned)(rows / 16));
  const dim3 blk(2 * Cpad);  // 32 * Cpad/16
  const size_t sh = shmem_bytes(K, C);
  if (Cpad == 64) {
    k_gemm<64><<<grid, blk, sh, s>>>(X, ldx, K, W, C, b, Y, ldy, act);
  } else if (Cpad == 32) {
    k_gemm<32><<<grid, blk, sh, s>>>(X, ldx, K, W, C, b, Y, ldy, act);
  } else {
    k_gemm<16><<<grid, blk, sh, s>>>(X, ldx, K, W, C, b, Y, ldy, act);
  }
}

static inline void gemm_gather(hipStream_t s, const float* D0, int w0,
                               const int* idxA, const int* idxB,
                               const float* XF, int K, long rows,
                               const float* W, int C, const float* b, float* Y,
                               int ldy, int act) {
  const dim3 grid((unsigned)(rows / 16));
  const dim3 blk(64);  // C=20 -> Cpad=32 -> 2 waves
  k_gemm_gather<32><<<grid, blk, shmem_bytes(K, C), s>>>(
      D0, w0, idxA, idxB, XF, K, W, C, b, Y, ldy, act);
}

extern "C" void kernel_launch(void* const* d_in, const int* in_sizes, int n_in,
                              void* d_out, int out_size, void* d_ws,
                              size_t ws_size, hipStream_t stream) {
  (void)n_in; (void)out_size; (void)ws_size;
  const float* x1 = (const float*)d_in[0];
  const float* x2 = (const float*)d_in[1];
  const int*   ei = (const int*)d_in[2];
  const float* attack   = (const float*)d_in[4];
  const int*   asrcs    = (const int*)d_in[5];
  const int*   adsts    = (const int*)d_in[6];
  const float* transfer = (const float*)d_in[7];
  const int*   tsrcs    = (const int*)d_in[8];
  const int*   tdsts    = (const int*)d_in[9];
  const float* deploy   = (const float*)d_in[10];
  const int*   dtgts    = (const int*)d_in[11];
  const int*   abtch    = (const int*)d_in[12];
  const int*   tbtch    = (const int*)d_in[13];
  const int*   dbtch    = (const int*)d_in[14];

  const float* atW  = (const float*)d_in[16]; const float* atb  = (const float*)d_in[17];
  const float* at2W = (const float*)d_in[18]; const float* at2b = (const float*)d_in[19];
  const float* Wg1  = (const float*)d_in[20]; const float* Wg2  = (const float*)d_in[21];
  const float* Wn1  = (const float*)d_in[22]; const float* Wn2  = (const float*)d_in[23];
  const float* bg1  = (const float*)d_in[24]; const float* bg2  = (const float*)d_in[25];
  const float* bn1  = (const float*)d_in[26]; const float* bn2  = (const float*)d_in[27];
  const float* dtW  = (const float*)d_in[28]; const float* dtb  = (const float*)d_in[29];
  const float* dt2W = (const float*)d_in[30]; const float* dt2b = (const float*)d_in[31];
  const float* folW = (const float*)d_in[32]; const float* folb = (const float*)d_in[33];
  const float* lin1W = (const float*)d_in[61]; const float* lin1b = (const float*)d_in[62];
  const float* lin2W = (const float*)d_in[63]; const float* lin2b = (const float*)d_in[64];
  const float* oaW  = (const float*)d_in[74]; const float* oab  = (const float*)d_in[75];
  const float* ttW  = (const float*)d_in[76]; const float* ttb  = (const float*)d_in[77];
  const float* tt2W = (const float*)d_in[78]; const float* tt2b = (const float*)d_in[79];

  const long N = in_sizes[0] / 15;
  const long B = in_sizes[1] / 4;
  const long E = in_sizes[2] / 2;
  const long NORD = in_sizes[5];
  const int* esrc = ei;
  const int* edst = ei + E;

  // workspace layout (floats)
  float* base = (float*)d_ws;
  size_t o = 0;
  auto alloc = [&](size_t nf) { float* p = base + o; o += nf; return p; };
  float* xfull = alloc((size_t)N * 165);
  float* bq    = alloc((size_t)N * GDIM);
  float* bk    = alloc((size_t)N * GDIM);   // bq..bk contiguous -> orders3
  float* bv    = alloc((size_t)N * GDIM);
  float* bs    = alloc((size_t)N * GDIM);   // bv..bs contiguous -> ordersO
  float* agg   = alloc((size_t)N * GDIM);
  float* tbuf  = alloc((size_t)N * GDIM);
  float* elog  = alloc((size_t)E);
  float* eexp  = alloc((size_t)E);
  unsigned* nmax = (unsigned*)alloc((size_t)N);
  float* nsum  = alloc((size_t)N);
  float* pooled = alloc((size_t)B * GDIM);
  float* orders3 = bq;   // [3*NORD, 20]
  float* ordersO = bv;   // [3*NORD, 20]
  float* hid     = agg;  // [NORD, 20]
  float* tmpBM   = tbuf; // [B*NM, 20]
  float* gl      = nsum; // gate logits [N] (reused after conv layers)
  float* Vout = (float*)d_out;
  float* Lout = (float*)d_out + B;

  const int TB = 256;
  auto cdiv = [](long a, long b) { return (unsigned)((a + b - 1) / b); };

  // xfull[:,0:15] = x1
  k_copy_x1<<<cdiv(N * 15, TB), TB, 0, stream>>>(x1, xfull, N * 15);

  // ---------------- 3 TransformerConv + GraphNorm layers ------------------
  const int Fin[3] = {15, 65, 115};
  for (int l = 0; l < 3; ++l) {
    const int bi = 34 + 9 * l;
    const float* Wbeta = (const float*)d_in[bi + 0];
    const float* Wk = (const float*)d_in[bi + 1];
    const float* Wq = (const float*)d_in[bi + 2];
    const float* Ws = (const float*)d_in[bi + 3];
    const float* Wv = (const float*)d_in[bi + 4];
    const float* bk_ = (const float*)d_in[bi + 5];
    const float* bq_ = (const float*)d_in[bi + 6];
    const float* bs_ = (const float*)d_in[bi + 7];
    const float* bv_ = (const float*)d_in[bi + 8];
    const int ni = 65 + 3 * l;
    const float* nbias = (const float*)d_in[ni + 0];
    const float* nms   = (const float*)d_in[ni + 1];
    const float* nwgt  = (const float*)d_in[ni + 2];

    const int K = Fin[l];
    gemm(stream, xfull, 165, K, N, Wq, GDIM, bq_, bq, GDIM, 0);
    gemm(stream, xfull, 165, K, N, Wk, GDIM, bk_, bk, GDIM, 0);
    gemm(stream, xfull, 165, K, N, Wv, GDIM, bv_, bv, GDIM, 0);
    gemm(stream, xfull, 165, K, N, Ws, GDIM, bs_, bs, GDIM, 0);

    k_fill<<<cdiv(N, TB), TB, 0, stream>>>((float*)nmax, 0.f, N);
    k_fill<<<cdiv(N, TB), TB, 0, stream>>>(nsum, 0.f, N);
    k_fill<<<cdiv(N * GDIM, TB), TB, 0, stream>>>(agg, 0.f, N * GDIM);

    k_edge_logits<<<cdiv(E, TB), TB, 0, stream>>>(esrc, edst, bq, bk, elog,
                                                  nmax, E);
    k_edge_exp<<<cdiv(E, TB), TB, 0, stream>>>(edst, elog, nmax, eexp, nsum, E);
    k_edge_agg<<<cdiv(E, TB), TB, 0, stream>>>(esrc, edst, eexp, nsum, bv, agg,
                                               E);

    k_beta<<<cdiv(N, TB), TB, 0, stream>>>(bs, agg, Wbeta, tbuf, N);
    k_gnorm<<<(unsigned)B, 256, 0, stream>>>(tbuf, nwgt, nbias, nms, xfull,
                                             15 + GDIM * l);
  }

  // ---------------- attention pooling + value head ------------------------
  gemm(stream, xfull, 165, 165, N, Wg1, GDIM, bg1, tbuf, GDIM, 2);   // leaky
  k_gate<<<cdiv(N, TB), TB, 0, stream>>>(tbuf, Wg2, bg2, gl, N);
  gemm(stream, xfull, 165, 165, N, Wn1, GDIM, bn1, agg, GDIM, 2);    // leaky
  gemm(stream, agg, GDIM, GDIM, N, Wn2, GDIM, bn2, bv, GDIM, 0);     // hn
  k_gate_pool<<<(unsigned)B, NPG, 0, stream>>>(gl, bv, pooled);
  k_value<<<(unsigned)B, 64, 0, stream>>>(pooled, x2, lin1W, lin1b, lin2W,
                                          lin2b, Vout);

  // ---------------- order MLPs -------------------------------------------
  gemm_gather(stream, attack, 22, asrcs, adsts, xfull, 352, NORD, atW, 20, atb,
              hid, 20, 1);
  gemm(stream, hid, 20, 20, NORD, at2W, 20, at2b, orders3, 20, 0);

  gemm_gather(stream, transfer, 21, tsrcs, tdsts, xfull, 351, NORD, ttW, 20,
              ttb, hid, 20, 1);
  gemm(stream, hid, 20, 20, NORD, tt2W, 20, tt2b, orders3 + NORD * 20, 20, 0);

  gemm_gather(stream, deploy, 11, dtgts, nullptr, xfull, 176, NORD, dtW, 20,
              dtb, hid, 20, 1);
  gemm(stream, hid, 20, 20, NORD, dt2W, 20, dt2b, orders3 + 2 * NORD * 20, 20,
       0);

  gemm(stream, orders3, 20, 20, 3 * NORD, oaW, 20, oab, ordersO, 20, 0);

  k_fill<<<cdiv(B * NM * 20, TB), TB, 0, stream>>>(tmpBM, 0.f, B * NM * 20);
  k_scatter<<<cdiv(3 * NORD, TB), TB, 0, stream>>>(ordersO, abtch, tbtch,
                                                   dbtch, NORD, tmpBM);
  k_fol_lsm<<<(unsigned)B, NM, 0, stream>>>(tmpBM, folW, folb, Lout);
}